// H2GCNConv_25555055411702
// MI455X (gfx1250) — compile-verified
//
#include <hip/hip_runtime.h>
#include <stdint.h>

#define D_FEAT 64
#define HCHUNK 1024   // ints per LDS staging buffer (4 KB)

// ---------------- CDNA5 async-to-LDS helpers (gfx1250, ASYNCcnt path) ----------------
__device__ __forceinline__ void async_copy_b128(uint32_t lds_addr, uint32_t byte_off,
                                                const void* base) {
  // GVS mode: mem = SGPR64 base + VGPR32 offset; VDST = per-lane LDS byte address
  asm volatile("global_load_async_to_lds_b128 %0, %1, %2"
               :: "v"(lds_addr), "v"(byte_off), "s"(base) : "memory");
}
__device__ __forceinline__ void wait_async0() { asm volatile("s_wait_asynccnt 0x0" ::: "memory"); }
__device__ __forceinline__ void wait_async1() { asm volatile("s_wait_asynccnt 0x1" ::: "memory"); }

// ---------------- zero helpers ----------------
__global__ __launch_bounds__(256) void zero_i32_kernel(int* __restrict__ p, int n) {
  int i = blockIdx.x * blockDim.x + threadIdx.x;
  if (i < n) p[i] = 0;
}
__global__ __launch_bounds__(256) void zero_f32_kernel(float* __restrict__ p, long n) {
  long i = (long)blockIdx.x * blockDim.x + threadIdx.x;
  if (i < n) p[i] = 0.0f;
}

// ---------------- 1) histogram of dst, edge stream staged via async DMA ----------------
__global__ __launch_bounds__(256) void hist_kernel(const int* __restrict__ dstIdx, int nEdges,
                                                   int* __restrict__ cnt) {
  __shared__ __align__(16) int sm[2][HCHUNK];
  const int tid = threadIdx.x;
  const int nFull = nEdges / HCHUNK;
  // low 32 bits of a flat pointer into the LDS aperture == LDS byte offset
  uint32_t ldsA[2];
  ldsA[0] = (uint32_t)(uintptr_t)(&sm[0][tid * 4]);
  ldsA[1] = (uint32_t)(uintptr_t)(&sm[1][tid * 4]);

  int chunk = blockIdx.x;
  int p = 0;
  if (chunk < nFull) {
    uint32_t voff = (uint32_t)(chunk * HCHUNK + tid * 4) * 4u;
    async_copy_b128(ldsA[0], voff, dstIdx);
  }
  while (chunk < nFull) {
    const int nxt = chunk + gridDim.x;          // uniform across block
    const bool pre = (nxt < nFull);
    if (pre) {
      uint32_t voff = (uint32_t)(nxt * HCHUNK + tid * 4) * 4u;
      async_copy_b128(ldsA[p ^ 1], voff, dstIdx);
    }
    if (pre) wait_async1(); else wait_async0(); // oldest (current buffer) complete
    __syncthreads();                            // all waves' DMA slices visible
#pragma unroll
    for (int k = 0; k < 4; ++k) {
      int d = sm[p][tid * 4 + k];
      atomicAdd(&cnt[d], 1);
    }
    __syncthreads();                            // done reading buf[p] before it refills
    p ^= 1;
    chunk = nxt;
  }
  // tail (non-multiple-of-HCHUNK remainder): plain guarded loads
  for (int i = nFull * HCHUNK + blockIdx.x * blockDim.x + tid; i < nEdges;
       i += gridDim.x * blockDim.x)
    atomicAdd(&cnt[dstIdx[i]], 1);
}

// ---------------- 2) single-block exclusive scan (wave32 shfl-based) ----------------
__global__ __launch_bounds__(1024) void scan_kernel(const int* __restrict__ cnt,
                                                    int* __restrict__ off, int n) {
  __shared__ int wsum[32];
  __shared__ int carry_sm;
  if (threadIdx.x == 0) carry_sm = 0;
  __syncthreads();
  const int lane = threadIdx.x & 31;
  const int wave = threadIdx.x >> 5;
  for (int base = 0; base < n; base += 1024) {
    const int i = base + threadIdx.x;
    const int v = (i < n) ? cnt[i] : 0;
    int sv = v;                                  // inclusive wave scan
#pragma unroll
    for (int d = 1; d < 32; d <<= 1) {
      int t = __shfl_up(sv, d);
      if (lane >= d) sv += t;
    }
    if (lane == 31) wsum[wave] = sv;
    __syncthreads();
    if (wave == 0) {                             // scan the 32 wave totals
      int ws = wsum[lane];
      int s2 = ws;
#pragma unroll
      for (int d = 1; d < 32; d <<= 1) {
        int t = __shfl_up(s2, d);
        if (lane >= d) s2 += t;
      }
      wsum[lane] = s2 - ws;                      // exclusive prefix of wave sums
    }
    __syncthreads();
    if (i < n) off[i] = carry_sm + wsum[wave] + (sv - v);
    __syncthreads();                             // all reads of carry_sm/wsum done
    if (threadIdx.x == 1023) carry_sm += wsum[31] + sv;  // chunk total
    __syncthreads();
  }
  if (threadIdx.x == 0) off[n] = carry_sm;
}

// ---------------- 3) reorder edges into per-row packed {w,src} ----------------
__global__ __launch_bounds__(256) void reorder_kernel(const int* __restrict__ dstIdx,
                                                      const int* __restrict__ srcIdx,
                                                      const float* __restrict__ w, int nEdges,
                                                      const int* __restrict__ off,
                                                      int* __restrict__ cur,
                                                      uint64_t* __restrict__ packed) {
  const int stride = gridDim.x * blockDim.x;
  for (int i = blockIdx.x * blockDim.x + threadIdx.x; i < nEdges; i += stride) {
    __builtin_prefetch(dstIdx + i + 8192, 0, 0);            // global_prefetch_b8, speculative
    const int d    = __builtin_nontemporal_load(dstIdx + i); // read-once streams: NT
    const int s    = __builtin_nontemporal_load(srcIdx + i);
    const float wt = __builtin_nontemporal_load(w + i);
    const int r = atomicAdd(&cur[d], 1);
    const uint64_t pk = ((uint64_t)__float_as_uint(wt) << 32) | (uint32_t)s;
    packed[off[d] + r] = pk;
  }
}

// ------- 4) fused atomic-free row gather for both hops: 16 lanes x float4 per row -------
__global__ __launch_bounds__(256) void gather_fused_kernel(const float* __restrict__ x,
                                                           const uint64_t* __restrict__ pk1,
                                                           const int* __restrict__ off1,
                                                           const uint64_t* __restrict__ pk2,
                                                           const int* __restrict__ off2,
                                                           float* __restrict__ out, int nNodes) {
  const int lane  = threadIdx.x & 15;
  const int group = (blockIdx.x * blockDim.x + threadIdx.x) >> 4;   // [0, 2N)
  if (group >= 2 * nNodes) return;
  const bool hop2 = (group >= nNodes);
  const int node = hop2 ? group - nNodes : group;
  const int* __restrict__ off = hop2 ? off2 : off1;
  const uint64_t* __restrict__ packed = hop2 ? pk2 : pk1;
  const int outColOfs = hop2 ? D_FEAT : 0;

  const int e0 = off[node], e1 = off[node + 1];
  __builtin_prefetch(packed + e0, 0, 0);         // pull row's edge list while we set up
  float4 acc = make_float4(0.f, 0.f, 0.f, 0.f);
  const float* xcol = x + lane * 4;

  int e = e0;
  for (; e + 1 < e1; e += 2) {                   // 2-way unroll: two rows in flight
    const uint64_t pa = packed[e];
    const uint64_t pb = packed[e + 1];
    const int   sa = (int)(uint32_t)pa;
    const int   sb = (int)(uint32_t)pb;
    const float wa = __uint_as_float((uint32_t)(pa >> 32));
    const float wb = __uint_as_float((uint32_t)(pb >> 32));
    const float4 va = *(const float4*)(xcol + (size_t)sa * D_FEAT);  // 256B coalesced, L2-hot
    const float4 vb = *(const float4*)(xcol + (size_t)sb * D_FEAT);
    acc.x += wa * va.x; acc.y += wa * va.y; acc.z += wa * va.z; acc.w += wa * va.w;
    acc.x += wb * vb.x; acc.y += wb * vb.y; acc.z += wb * vb.z; acc.w += wb * vb.w;
  }
  if (e < e1) {
    const uint64_t pa = packed[e];
    const int   sa = (int)(uint32_t)pa;
    const float wa = __uint_as_float((uint32_t)(pa >> 32));
    const float4 va = *(const float4*)(xcol + (size_t)sa * D_FEAT);
    acc.x += wa * va.x; acc.y += wa * va.y; acc.z += wa * va.z; acc.w += wa * va.w;
  }
  *(float4*)(out + (size_t)node * (2 * D_FEAT) + outColOfs + lane * 4) = acc;
}

// ---------------- fallback: direct atomic scatter (if scratch too small) ----------------
__global__ __launch_bounds__(256) void spmm_atomic_kernel(const float* __restrict__ x,
                                                          const int* __restrict__ dstIdx,
                                                          const int* __restrict__ srcIdx,
                                                          const float* __restrict__ w, int nEdges,
                                                          float* __restrict__ out, int outColOfs) {
  const long t = (long)blockIdx.x * blockDim.x + threadIdx.x;
  const int lane = (int)(t & 15);
  long e = t >> 4;
  const long stride = ((long)gridDim.x * blockDim.x) >> 4;
  for (; e < nEdges; e += stride) {
    const int d = dstIdx[e], s = srcIdx[e];
    const float wt = w[e];
    const float4 v = *(const float4*)(x + (size_t)s * D_FEAT + lane * 4);
    float* o = out + (size_t)d * (2 * D_FEAT) + outColOfs + lane * 4;
    atomicAdd(o + 0, wt * v.x); atomicAdd(o + 1, wt * v.y);
    atomicAdd(o + 2, wt * v.z); atomicAdd(o + 3, wt * v.w);
  }
}

extern "C" void kernel_launch(void* const* d_in, const int* in_sizes, int n_in,
                              void* d_out, int out_size, void* d_ws, size_t ws_size,
                              hipStream_t stream) {
  (void)n_in; (void)out_size;
  const float* x  = (const float*)d_in[0];
  const int*   ei1 = (const int*)d_in[1];
  const float* w1 = (const float*)d_in[2];
  const int*   ei2 = (const int*)d_in[3];
  const float* w2 = (const float*)d_in[4];
  const int N  = in_sizes[0] / D_FEAT;
  const int E1 = in_sizes[2];
  const int E2 = in_sizes[4];
  const int* dst1 = ei1;  const int* src1 = ei1 + E1;   // edge_index[0], edge_index[1]
  const int* dst2 = ei2;  const int* src2 = ei2 + E2;
  float* out = (float*)d_out;

  const size_t need = (size_t)E1 * 8 + (size_t)E2 * 8 + (size_t)(2 * (N + 1) + 2 * N) * 4;
  if (ws_size >= need) {
    char* p = (char*)d_ws;
    uint64_t* pk1 = (uint64_t*)p; p += (size_t)E1 * 8;
    uint64_t* pk2 = (uint64_t*)p; p += (size_t)E2 * 8;
    int* off1 = (int*)p; p += (size_t)(N + 1) * 4;
    int* off2 = (int*)p; p += (size_t)(N + 1) * 4;
    int* cnt1 = (int*)p; p += (size_t)N * 4;
    int* cnt2 = (int*)p;                                  // cnt1,cnt2 contiguous

    zero_i32_kernel<<<(2 * N + 255) / 256, 256, 0, stream>>>(cnt1, 2 * N);
    {
      int nf1 = E1 / HCHUNK; int b1 = nf1 > 0 ? (nf1 < 2048 ? nf1 : 2048) : 1;
      hist_kernel<<<b1, 256, 0, stream>>>(dst1, E1, cnt1);
      int nf2 = E2 / HCHUNK; int b2 = nf2 > 0 ? (nf2 < 2048 ? nf2 : 2048) : 1;
      hist_kernel<<<b2, 256, 0, stream>>>(dst2, E2, cnt2);
    }
    scan_kernel<<<1, 1024, 0, stream>>>(cnt1, off1, N);
    scan_kernel<<<1, 1024, 0, stream>>>(cnt2, off2, N);
    zero_i32_kernel<<<(2 * N + 255) / 256, 256, 0, stream>>>(cnt1, 2 * N);
    reorder_kernel<<<(E1 + 255) / 256, 256, 0, stream>>>(dst1, src1, w1, E1, off1, cnt1, pk1);
    reorder_kernel<<<(E2 + 255) / 256, 256, 0, stream>>>(dst2, src2, w2, E2, off2, cnt2, pk2);
    const int gb = (2 * N * 16 + 255) / 256;              // both hops in one grid
    gather_fused_kernel<<<gb, 256, 0, stream>>>(x, pk1, off1, pk2, off2, out, N);
  } else {
    const long total = (long)N * 2 * D_FEAT;
    zero_f32_kernel<<<(int)((total + 255) / 256), 256, 0, stream>>>(out, total);
    const long t1 = (long)E1 * 16;
    spmm_atomic_kernel<<<(int)((t1 + 255) / 256), 256, 0, stream>>>(x, dst1, src1, w1, E1, out, 0);
    const long t2 = (long)E2 * 16;
    spmm_atomic_kernel<<<(int)((t2 + 255) / 256), 256, 0, stream>>>(x, dst2, src2, w2, E2, out, D_FEAT);
  }
}